// DGCNN_Grouper_23175643529472
// MI455X (gfx1250) — compile-verified
//
#include <hip/hip_runtime.h>
#include <hip/hip_bf16.h>

typedef __attribute__((ext_vector_type(2))) float v2f;
typedef __attribute__((ext_vector_type(8))) float v8f;

#define NB 4          // batch
#define NPTS 4096     // initial points
#define MPTS 1024     // downsampled points
#define KNN 16        // neighbors
#define GROUPS 4

#if defined(__has_builtin)
#if __has_builtin(__builtin_amdgcn_global_load_async_to_lds_b32)
#define HAVE_ASYNC_LDS 1
#endif
#endif

typedef __attribute__((address_space(1))) int* gptr_i32;
typedef __attribute__((address_space(3))) int* lptr_i32;

// ---------------------------------------------------------------------------
// 0) transpose input + initial 3->8 channel embedding  f0 = Wi@coor + bi
// ---------------------------------------------------------------------------
__global__ void k_init(const float* __restrict__ x, const float* __restrict__ Wi,
                       const float* __restrict__ bi, float* __restrict__ coor,
                       float* __restrict__ normal, float* __restrict__ plane,
                       float* __restrict__ f0, int N) {
  int t = blockIdx.x * blockDim.x + threadIdx.x;
  if (t >= NB * N) return;
  int n = t % N, b = t / N;
  const float* xp = x + ((size_t)b * N + n) * 7;
  float cx = xp[0], cy = xp[1], cz = xp[2];
  coor[((size_t)b * 3 + 0) * N + n] = cx;
  coor[((size_t)b * 3 + 1) * N + n] = cy;
  coor[((size_t)b * 3 + 2) * N + n] = cz;
  normal[((size_t)b * 3 + 0) * N + n] = xp[3];
  normal[((size_t)b * 3 + 1) * N + n] = xp[4];
  normal[((size_t)b * 3 + 2) * N + n] = xp[5];
  plane[(size_t)b * N + n] = xp[6];
#pragma unroll
  for (int o = 0; o < 8; ++o)
    f0[((size_t)b * 8 + o) * N + n] =
        Wi[o * 3 + 0] * cx + Wi[o * 3 + 1] * cy + Wi[o * 3 + 2] * cz + bi[o];
}

// ---------------------------------------------------------------------------
// 1) KNN: one wave owns 16 queries; 16x16 distance tiles via WMMA f32 16x16x4
//    d = |q|^2 + |k|^2 - 2 q.k  (A = queries 16x4, B = keys 4x16)
// ---------------------------------------------------------------------------
__global__ void __launch_bounds__(128)
k_knn(const float* __restrict__ coorQ, const float* __restrict__ coorK,
      int* __restrict__ knn_idx, int Nq, int Nk) {
  __shared__ float s_q2[4][16];
  __shared__ float s_tile[4][16][16];
  const int wave = threadIdx.x >> 5;
  const int lane = threadIdx.x & 31;
  const int half = lane >> 4;
  const int l16 = lane & 15;
  const int b = blockIdx.y;
  const int qt = blockIdx.x * 4 + wave;            // query tile (16 queries)
  const int q = qt * 16 + l16;

  const float* cq = coorQ + (size_t)b * 3 * Nq;
  const float qx = cq[0 * Nq + q], qy = cq[1 * Nq + q], qz = cq[2 * Nq + q];
  v2f afrag;                                       // A: lanes0-15 K=0,1 ; lanes16-31 K=2,3
  afrag.x = half ? qz : qx;
  afrag.y = half ? 0.0f : qy;
  if (half == 0) s_q2[wave][l16] = qx * qx + qy * qy + qz * qz;

  float topd[16];
  int topi[16];
#pragma unroll
  for (int i = 0; i < 16; ++i) { topd[i] = 3.0e38f; topi[i] = 0; }

  const float* ck = coorK + (size_t)b * 3 * Nk;
  for (int kb = 0; kb < Nk; kb += 16) {
    const int kidx = kb + l16;
    const float kx = ck[0 * Nk + kidx], ky = ck[1 * Nk + kidx], kz = ck[2 * Nk + kidx];
    if (kb + 16 < Nk) __builtin_prefetch(&ck[kb + 16 + l16], 0, 1);
    v2f bfrag;                                     // B mirrored layout: lanes=N, K=2*half+{0,1}
    bfrag.x = half ? kz : kx;
    bfrag.y = half ? 0.0f : ky;
    const float k2 = kx * kx + ky * ky + kz * kz;
    v8f acc = {};
    acc = __builtin_amdgcn_wmma_f32_16x16x4_f32(false, afrag, false, bfrag,
                                                (short)0, acc, false, false);
    // D: vgpr r -> M = r + 8*half, N = l16
#pragma unroll
    for (int r = 0; r < 8; ++r) {
      const int m = r + 8 * half;
      s_tile[wave][m][l16] = s_q2[wave][m] + k2 - 2.0f * acc[r];
    }
    if (half == 0) {                               // lane l16 owns query row l16
#pragma unroll 1
      for (int j = 0; j < 16; ++j) {
        float cd = s_tile[wave][l16][j];
        int ci = kb + j;
        if (cd < topd[15]) {
#pragma unroll
          for (int t = 0; t < 16; ++t) {           // sorted swap-chain insert
            const bool sw = cd < topd[t];
            const float td = sw ? topd[t] : cd;
            const int ti = sw ? topi[t] : ci;
            topd[t] = sw ? cd : topd[t];
            topi[t] = sw ? ci : topi[t];
            cd = td; ci = ti;
          }
        }
      }
    }
  }
  if (half == 0) {
    int* op = knn_idx + ((size_t)b * Nq + q) * KNN;
#pragma unroll
    for (int t = 0; t < 16; ++t) op[t] = topi[t];
  }
}

// ---------------------------------------------------------------------------
// 2) fused graph-feature + GEMM: h[oc,col] = sum_c W[oc,c] * f4[c,col]
//    f4[c,(nq,kk)] = c<C ? fK[c,idx]-fQ[c,nq] : fQ[c-C,nq]
//    WMMA f32 16x16x4, wave computes a 16(oc) x 16(col) tile.
//    A 16-wide column tile is exactly one query point: nq is wave-uniform, so
//    center-feature reads become scalar loads. The 16xC2 weight tile is staged
//    once per block into LDS via async global->LDS loads (ASYNCcnt).
// ---------------------------------------------------------------------------
__global__ void __launch_bounds__(128)
k_gemm(const float* __restrict__ W, const float* __restrict__ fK,
       const float* __restrict__ fQ, const int* __restrict__ idx,
       float* __restrict__ h, int C, int OC, int NQ, int NK) {
  __shared__ float sW[16 * 128];                   // 16 out-ch x (2C<=128) weights
  const int wave = threadIdx.x >> 5;
  const int lane = threadIdx.x & 31;
  const int half = lane >> 4;
  const int l16 = lane & 15;
  const int b = blockIdx.z;
  const int ocb = blockIdx.y * 16;
  const int C2 = 2 * C;

  // stage weight tile W[ocb:ocb+16, :] into LDS (async to LDS when available)
  {
    const float* Wg = W + (size_t)ocb * C2;
    const int elems = 16 * C2;
    for (int i = threadIdx.x; i < elems; i += 128) {
#ifdef HAVE_ASYNC_LDS
      __builtin_amdgcn_global_load_async_to_lds_b32(
          (gptr_i32)(Wg + i), (lptr_i32)(sW + i), 0, 0);
#else
      sW[i] = Wg[i];
#endif
    }
#ifdef HAVE_ASYNC_LDS
#if __has_builtin(__builtin_amdgcn_s_wait_asynccnt)
    __builtin_amdgcn_s_wait_asynccnt(0);
#else
    asm volatile("s_wait_asynccnt 0x0" ::: "memory");
#endif
#endif
    __syncthreads();
  }

  const int nq = blockIdx.x * 4 + wave;            // one query per 16-col tile
  const int col = nq * 16 + l16;                   // kk == l16
  const int cols = NQ * KNN;
  const int id = idx[((size_t)b * NQ + nq) * KNN + l16];
  const float* fKb = fK + (size_t)b * C * NK;
  const float* fQb = fQ + (size_t)b * C * NQ;

  v8f acc = {};
  for (int c = 0; c < C2; c += 4) {
    const int cc0 = c + 2 * half;
    const int cc1 = cc0 + 1;
    v2f a, bb;
    a.x = sW[l16 * C2 + cc0];
    a.y = sW[l16 * C2 + cc1];
    bb.x = (cc0 < C) ? (fKb[(size_t)cc0 * NK + id] - fQb[(size_t)cc0 * NQ + nq])
                     : fQb[(size_t)(cc0 - C) * NQ + nq];
    bb.y = (cc1 < C) ? (fKb[(size_t)cc1 * NK + id] - fQb[(size_t)cc1 * NQ + nq])
                     : fQb[(size_t)(cc1 - C) * NQ + nq];
    acc = __builtin_amdgcn_wmma_f32_16x16x4_f32(false, a, false, bb,
                                                (short)0, acc, false, false);
  }
  float* hb = h + (size_t)b * OC * cols;
#pragma unroll
  for (int r = 0; r < 8; ++r)
    hb[(size_t)(ocb + r + 8 * half) * cols + col] = acc[r];
}

// ---------------------------------------------------------------------------
// 3) GroupNorm stats: sum/sumsq per (b,group), group region is contiguous in h
// ---------------------------------------------------------------------------
__global__ void k_stats(const float* __restrict__ h, float* __restrict__ stats,
                        int OC, int cols) {
  const int bg = blockIdx.y;                 // b*GROUPS + g
  const int ocg = OC / GROUPS;
  const int b = bg / GROUPS, g = bg % GROUPS;
  const float* base = h + ((size_t)b * OC + (size_t)g * ocg) * cols;
  const size_t n = (size_t)ocg * cols;
  float s = 0.f, s2 = 0.f;
  for (size_t i = (size_t)blockIdx.x * blockDim.x + threadIdx.x; i < n;
       i += (size_t)gridDim.x * blockDim.x) {
    const float v = base[i];
    s += v; s2 += v * v;
  }
  __shared__ float rs[256], rq[256];
  rs[threadIdx.x] = s; rq[threadIdx.x] = s2;
  __syncthreads();
  for (int off = 128; off > 0; off >>= 1) {
    if (threadIdx.x < off) {
      rs[threadIdx.x] += rs[threadIdx.x + off];
      rq[threadIdx.x] += rq[threadIdx.x + off];
    }
    __syncthreads();
  }
  if (threadIdx.x == 0) {
    atomicAdd(&stats[2 * bg + 0], rs[0]);
    atomicAdd(&stats[2 * bg + 1], rq[0]);
  }
}

// ---------------------------------------------------------------------------
// 4) normalize + leaky-relu + max over K
// ---------------------------------------------------------------------------
__global__ void k_gn_max(const float* __restrict__ h, const float* __restrict__ stats,
                         const float* __restrict__ gamma, const float* __restrict__ beta,
                         float* __restrict__ fout, int OC, int NQ) {
  const int t = blockIdx.x * blockDim.x + threadIdx.x;
  if (t >= NB * OC * NQ) return;
  const int nq = t % NQ;
  const int oc = (t / NQ) % OC;
  const int b = t / (NQ * OC);
  const int g = oc / (OC / GROUPS);
  const float cnt = (float)(OC / GROUPS) * (float)NQ * (float)KNN;
  const float mu = stats[2 * (b * GROUPS + g) + 0] / cnt;
  const float var = stats[2 * (b * GROUPS + g) + 1] / cnt - mu * mu;
  const float rs = rsqrtf(var + 1e-5f);
  const float ga = gamma[oc], be = beta[oc];
  const int cols = NQ * KNN;
  const float* hp = h + ((size_t)b * OC + oc) * cols + (size_t)nq * KNN;
  float m = -3.0e38f;
#pragma unroll
  for (int k = 0; k < KNN; ++k) {
    float v = (hp[k] - mu) * rs * ga + be;
    v = v > 0.f ? v : 0.2f * v;
    m = fmaxf(m, v);
  }
  fout[((size_t)b * OC + oc) * NQ + nq] = m;
}

// ---------------------------------------------------------------------------
// 5) farthest point sampling: one workgroup per batch, dists in LDS
// ---------------------------------------------------------------------------
__global__ void __launch_bounds__(256)
k_fps(const float* __restrict__ coor, int N, int n_samp, int* __restrict__ out_idx) {
  __shared__ float s_dist[NPTS];
  __shared__ float s_rv[256];
  __shared__ int s_ri[256];
  __shared__ int s_last;
  const int b = blockIdx.x;
  const int tid = threadIdx.x;
  const float* cx = coor + (size_t)b * 3 * N;
  const float* cy = cx + N;
  const float* cz = cx + 2 * N;
  for (int i = tid; i < N; i += 256) s_dist[i] = 1e10f;
  if (tid == 0) s_last = 0;
  __syncthreads();
  for (int it = 0; it < n_samp; ++it) {
    const int last = s_last;
    if (tid == 0) out_idx[b * n_samp + it] = last;
    const float lx = cx[last], ly = cy[last], lz = cz[last];
    float bestv = -1.f;
    int besti = 0;
    for (int i = tid; i < N; i += 256) {
      const float dx = cx[i] - lx, dy = cy[i] - ly, dz = cz[i] - lz;
      const float d = dx * dx + dy * dy + dz * dz;
      const float dm = fminf(s_dist[i], d);
      s_dist[i] = dm;
      if (dm > bestv) { bestv = dm; besti = i; }
    }
    s_rv[tid] = bestv; s_ri[tid] = besti;
    __syncthreads();
    for (int off = 128; off > 0; off >>= 1) {
      if (tid < off && s_rv[tid + off] > s_rv[tid]) {
        s_rv[tid] = s_rv[tid + off];
        s_ri[tid] = s_ri[tid + off];
      }
      __syncthreads();
    }
    if (tid == 0) s_last = s_ri[0];
    __syncthreads();
  }
}

// ---------------------------------------------------------------------------
// 6) gather coords/normals/plane/features by FPS indices
// ---------------------------------------------------------------------------
__global__ void k_gather(const float* __restrict__ coor, const float* __restrict__ normal,
                         const float* __restrict__ plane, const float* __restrict__ fin,
                         const int* __restrict__ fidx, float* __restrict__ coor_q,
                         float* __restrict__ normal_q, float* __restrict__ plane_q,
                         float* __restrict__ fq, int N, int M, int C) {
  const int t = blockIdx.x * blockDim.x + threadIdx.x;
  if (t >= NB * M) return;
  const int m = t % M, b = t / M;
  const int id = fidx[b * M + m];
#pragma unroll
  for (int c = 0; c < 3; ++c) {
    coor_q[((size_t)b * 3 + c) * M + m] = coor[((size_t)b * 3 + c) * N + id];
    normal_q[((size_t)b * 3 + c) * M + m] = normal[((size_t)b * 3 + c) * N + id];
  }
  plane_q[(size_t)b * M + m] = plane[(size_t)b * N + id];
  for (int c = 0; c < C; ++c)
    fq[((size_t)b * C + c) * M + m] = fin[((size_t)b * C + c) * N + id];
}

// ---------------------------------------------------------------------------
// 7) pack outputs: coor^T (B,M,3) | f^T (B,M,128) | normal^T (B,M,3) | plane^T (B,M,1)
// ---------------------------------------------------------------------------
__global__ void k_finalize(const float* __restrict__ coor_q, const float* __restrict__ fin,
                           const float* __restrict__ normal_q, const float* __restrict__ plane_q,
                           float* __restrict__ out, int M) {
  const int t = blockIdx.x * blockDim.x + threadIdx.x;
  if (t >= NB * M) return;
  const int m = t % M, b = t / M;
  float* o1 = out;
  float* o2 = o1 + (size_t)NB * M * 3;
  float* o3 = o2 + (size_t)NB * M * 128;
  float* o4 = o3 + (size_t)NB * M * 3;
#pragma unroll
  for (int c = 0; c < 3; ++c) {
    o1[((size_t)b * M + m) * 3 + c] = coor_q[((size_t)b * 3 + c) * M + m];
    o3[((size_t)b * M + m) * 3 + c] = normal_q[((size_t)b * 3 + c) * M + m];
  }
  for (int c = 0; c < 128; ++c)
    o2[((size_t)b * M + m) * 128 + c] = fin[((size_t)b * 128 + c) * M + m];
  o4[(size_t)b * M + m] = plane_q[(size_t)b * M + m];
}

// ---------------------------------------------------------------------------
static inline float* carveF(char*& p, size_t nfloats) {
  float* r = (float*)p;
  p += ((nfloats * sizeof(float) + 255) / 256) * 256;
  return r;
}
static inline int* carveI(char*& p, size_t nints) {
  int* r = (int*)p;
  p += ((nints * sizeof(int) + 255) / 256) * 256;
  return r;
}

extern "C" void kernel_launch(void* const* d_in, const int* in_sizes, int n_in,
                              void* d_out, int out_size, void* d_ws, size_t ws_size,
                              hipStream_t stream) {
  const float* x = (const float*)d_in[0];
  const float* Wi = (const float*)d_in[2];
  const float* bi = (const float*)d_in[3];
  const float* W1 = (const float*)d_in[4];
  const float* g1 = (const float*)d_in[5];
  const float* b1 = (const float*)d_in[6];
  const float* W2 = (const float*)d_in[7];
  const float* g2 = (const float*)d_in[8];
  const float* b2 = (const float*)d_in[9];
  const float* W3 = (const float*)d_in[10];
  const float* g3 = (const float*)d_in[11];
  const float* b3 = (const float*)d_in[12];
  const float* W4 = (const float*)d_in[13];
  const float* g4 = (const float*)d_in[14];
  const float* b4 = (const float*)d_in[15];
  float* out = (float*)d_out;

  const int N = NPTS, M = MPTS;
  char* p = (char*)d_ws;
  float* coor = carveF(p, (size_t)NB * 3 * N);
  float* normal = carveF(p, (size_t)NB * 3 * N);
  float* plane = carveF(p, (size_t)NB * N);
  float* f0 = carveF(p, (size_t)NB * 8 * N);
  float* fA = carveF(p, (size_t)NB * 128 * N);
  float* fB = carveF(p, (size_t)NB * 128 * N);
  float* fQ2 = carveF(p, (size_t)NB * 128 * M);
  float* hbuf = carveF(p, (size_t)NB * 128 * M * KNN);  // == NB*32*N*KNN
  float* coor_q = carveF(p, (size_t)NB * 3 * M);
  float* normal_q = carveF(p, (size_t)NB * 3 * M);
  float* plane_q = carveF(p, (size_t)NB * M);
  float* coor_q2 = carveF(p, (size_t)NB * 3 * M);
  float* normal_q2 = carveF(p, (size_t)NB * 3 * M);
  float* plane_q2 = carveF(p, (size_t)NB * M);
  float* stats = carveF(p, 2 * NB * GROUPS);
  int* idx = carveI(p, (size_t)NB * N * KNN);
  int* fpsidx = carveI(p, (size_t)NB * M);

  // init: transpose + 3->8 embedding
  k_init<<<(NB * N + 255) / 256, 256, 0, stream>>>(x, Wi, bi, coor, normal, plane, f0, N);

  // ---- stage 1: knn(4096x4096), gemm 16->32, GN, maxK -> fA (B,32,N)
  k_knn<<<dim3((N / 16) / 4, NB), 128, 0, stream>>>(coor, coor, idx, N, N);
  k_gemm<<<dim3(N / 4, 32 / 16, NB), 128, 0, stream>>>(W1, f0, f0, idx, hbuf, 8, 32, N, N);
  (void)hipMemsetAsync(stats, 0, 2 * NB * GROUPS * sizeof(float), stream);
  k_stats<<<dim3(64, NB * GROUPS), 256, 0, stream>>>(hbuf, stats, 32, N * KNN);
  k_gn_max<<<(NB * 32 * N + 255) / 256, 256, 0, stream>>>(hbuf, stats, g1, b1, fA, 32, N);

  // ---- FPS 4096 -> 1024 and gather (features 32ch -> fB)
  k_fps<<<NB, 256, 0, stream>>>(coor, N, M, fpsidx);
  k_gather<<<(NB * M + 255) / 256, 256, 0, stream>>>(coor, normal, plane, fA, fpsidx,
                                                     coor_q, normal_q, plane_q, fB, N, M, 32);

  // ---- stage 2: knn(1024 q x 4096 k), gemm 64->64 -> fA (B,64,M)
  k_knn<<<dim3((M / 16) / 4, NB), 128, 0, stream>>>(coor_q, coor, idx, M, N);
  k_gemm<<<dim3(M / 4, 64 / 16, NB), 128, 0, stream>>>(W2, fA, fB, idx, hbuf, 32, 64, M, N);
  (void)hipMemsetAsync(stats, 0, 2 * NB * GROUPS * sizeof(float), stream);
  k_stats<<<dim3(64, NB * GROUPS), 256, 0, stream>>>(hbuf, stats, 64, M * KNN);
  k_gn_max<<<(NB * 64 * M + 255) / 256, 256, 0, stream>>>(hbuf, stats, g2, b2, fA, 64, M);

  // ---- stage 3: knn(1024x1024), gemm 128->64 -> fB (B,64,M)
  k_knn<<<dim3((M / 16) / 4, NB), 128, 0, stream>>>(coor_q, coor_q, idx, M, M);
  k_gemm<<<dim3(M / 4, 64 / 16, NB), 128, 0, stream>>>(W3, fA, fA, idx, hbuf, 64, 64, M, M);
  (void)hipMemsetAsync(stats, 0, 2 * NB * GROUPS * sizeof(float), stream);
  k_stats<<<dim3(64, NB * GROUPS), 256, 0, stream>>>(hbuf, stats, 64, M * KNN);
  k_gn_max<<<(NB * 64 * M + 255) / 256, 256, 0, stream>>>(hbuf, stats, g3, b3, fB, 64, M);

  // ---- FPS 1024 -> 1024 (FPS permutation) and gather (features 64ch -> fQ2)
  k_fps<<<NB, 256, 0, stream>>>(coor_q, M, M, fpsidx);
  k_gather<<<(NB * M + 255) / 256, 256, 0, stream>>>(coor_q, normal_q, plane_q, fB, fpsidx,
                                                     coor_q2, normal_q2, plane_q2, fQ2, M, M, 64);

  // ---- stage 4: knn(1024 q2 x 1024 k), gemm 128->128 -> fA (B,128,M)
  k_knn<<<dim3((M / 16) / 4, NB), 128, 0, stream>>>(coor_q2, coor_q, idx, M, M);
  k_gemm<<<dim3(M / 4, 128 / 16, NB), 128, 0, stream>>>(W4, fB, fQ2, idx, hbuf, 64, 128, M, M);
  (void)hipMemsetAsync(stats, 0, 2 * NB * GROUPS * sizeof(float), stream);
  k_stats<<<dim3(64, NB * GROUPS), 256, 0, stream>>>(hbuf, stats, 128, M * KNN);
  k_gn_max<<<(NB * 128 * M + 255) / 256, 256, 0, stream>>>(hbuf, stats, g4, b4, fA, 128, M);

  // ---- pack outputs
  k_finalize<<<(NB * M + 255) / 256, 256, 0, stream>>>(coor_q2, fA, normal_q2, plane_q2, out, M);
}